// ZFCSHGCNCNN_62354335203837
// MI455X (gfx1250) — compile-verified
//
#include <hip/hip_runtime.h>
#include <hip/hip_bf16.h>

// ---------------------------------------------------------------------------
// Dims (fixed by the reference): B=16, N=2000, E=3000, C=17, W=12, D=16,
// O=256, K=3 (diffusion), NB=3.
// ---------------------------------------------------------------------------
#define NB_B   16
#define NB_N   2000
#define NB_E   3000
#define NB_C   17
#define NB_W   12
#define NB_D   16
#define NB_O   256
#define NB_KC  51        // K*C = 3*17
#define NB_O4  64        // O/4
#define NB_BC  272       // B*C

typedef float v2f __attribute__((ext_vector_type(2)));
typedef float v8f __attribute__((ext_vector_type(8)));

// ---------------------------------------------------------------------------
// K-loop body for one 16x16 output tile, V_WMMA_F32_16X16X4_F32, unrolled x2.
// MASKED=false is the fast path (full tile): loads + pointer adds + wmma only.
// MASKED=true multiplies the streams by 0/1 lane masks (v_cndmask-generated,
// no EXEC divergence — WMMA requires EXEC all-ones).
// ---------------------------------------------------------------------------
template <bool MASKED>
__device__ __forceinline__ v8f gemm_kloop(
    const float* __restrict__ pa,   // A stream: this lane's row, +2*half
    const float* __restrict__ pb0,  // B stream row (2*half)
    const float* __restrict__ pb1,  // B stream row (2*half + 1)
    long stepB, int Kdim, float am, float bmask)
{
    v8f acc = {0.f, 0.f, 0.f, 0.f, 0.f, 0.f, 0.f, 0.f};
    const int kmain = Kdim & ~7;          // multiple of 8
    for (int k = 0; k < kmain; k += 8) {
        if (k + 72 < Kdim) {
            // keep the A stream ahead in cache (emits global_prefetch_b8)
            __builtin_prefetch(pa + 64, 0, 1);
        }
        v2f a0, b0, a1, b1;
        if (MASKED) {
            a0.x = pa[0] * am;      a0.y = pa[1] * am;
            b0.x = pb0[0] * bmask;  b0.y = pb1[0] * bmask;
            a1.x = pa[4] * am;      a1.y = pa[5] * am;
            b1.x = pb0[stepB] * bmask;
            b1.y = pb1[stepB] * bmask;
        } else {
            a0.x = pa[0];           a0.y = pa[1];
            b0.x = pb0[0];          b0.y = pb1[0];
            a1.x = pa[4];           a1.y = pa[5];
            b1.x = pb0[stepB];      b1.y = pb1[stepB];
        }
        // 8 args: (neg_a, A, neg_b, B, c_mod, C, reuse_a, reuse_b)
        acc = __builtin_amdgcn_wmma_f32_16x16x4_f32(
                  false, a0, false, b0, (short)0, acc, false, false);
        acc = __builtin_amdgcn_wmma_f32_16x16x4_f32(
                  false, a1, false, b1, (short)0, acc, false, false);
        pa  += 8;
        pb0 += 2 * stepB;
        pb1 += 2 * stepB;
    }
    if (kmain < Kdim) {                   // single 4-K tail (K % 8 == 4)
        v2f a0, b0;
        if (MASKED) {
            a0.x = pa[0] * am;      a0.y = pa[1] * am;
            b0.x = pb0[0] * bmask;  b0.y = pb1[0] * bmask;
        } else {
            a0.x = pa[0];           a0.y = pa[1];
            b0.x = pb0[0];          b0.y = pb1[0];
        }
        acc = __builtin_amdgcn_wmma_f32_16x16x4_f32(
                  false, a0, false, b0, (short)0, acc, false, false);
    }
    return acc;
}

// ---------------------------------------------------------------------------
// Generic strided fp32 WMMA GEMM:  C[M x Ncols] = A[M x K] * B[K x Ncols]
//   A(m,k) = A[m*lda + k]            (row-major, contiguous k)
//   B(k,c) = B[k*ldbk + c*ldbc]      (ldbc=1: row-major; ldbk=1: B^T)
// One wave per 16x16 output tile.  Full tiles (the overwhelmingly common
// case) take a mask-free loop selected by a *wave-uniform* branch, so EXEC
// stays all-ones through every WMMA on both paths.  K must be % 4 == 0.
// ---------------------------------------------------------------------------
__global__ __launch_bounds__(32) void k_wmma_gemm_f32(
    const float* __restrict__ A, long lda,
    const float* __restrict__ Bm, long ldbk, long ldbc,
    float* __restrict__ Cm, long ldc,
    int M, int Ncols, int Kdim)
{
    const int lane = threadIdx.x & 31;
    const int half = lane >> 4;        // 0: K pair (0,1)   1: K pair (2,3)
    const int l    = lane & 15;
    const int mt   = blockIdx.x * 16;
    const int nt   = blockIdx.y * 16;

    const int  arow = mt + l;
    const int  bcol = nt + l;
    const bool aok  = (arow < M);
    const bool bok  = (bcol < Ncols);
    const long ar   = aok ? (long)arow : 0;   // clamped: loads stay in-bounds
    const long bc   = bok ? (long)bcol : 0;

    const float* __restrict__ pa  = A  + ar * lda + 2 * half;
    const float* __restrict__ pb0 = Bm + (long)(2 * half) * ldbk + bc * ldbc;
    const float* __restrict__ pb1 = pb0 + ldbk;
    const long  stepB = 4 * ldbk;     // floats per 4-K advance of the B rows

    const bool fullTile = (mt + 16 <= M) && (nt + 16 <= Ncols);  // wave-uniform

    v8f acc;
    if (fullTile) {
        acc = gemm_kloop<false>(pa, pb0, pb1, stepB, Kdim, 1.f, 1.f);
    } else {
        const float am    = aok ? 1.0f : 0.0f;
        const float bmask = bok ? 1.0f : 0.0f;
        acc = gemm_kloop<true>(pa, pb0, pb1, stepB, Kdim, am, bmask);
    }

#pragma unroll
    for (int v = 0; v < 8; ++v) {
        const int row = mt + v + 8 * half;   // C/D layout: VGPR v -> rows v / v+8
        if (row < M && bcol < Ncols)
            Cm[(long)row * ldc + bcol] = acc[v];
    }
}

// ---------------------------------------------------------------------------
// S = softmax(relu(ne @ ne^T) with diag := stay_cost), one row per workgroup.
// (The reference's trailing S / rowsum(S) is an exact no-op after softmax.)
// ---------------------------------------------------------------------------
__global__ __launch_bounds__(256) void k_compute_S(
    const float* __restrict__ ne, const int* __restrict__ stay_cost,
    float* __restrict__ S)
{
    __shared__ float red[256];
    __shared__ float neL[NB_D];
    const int i   = blockIdx.x;
    const int tid = threadIdx.x;
    if (tid < NB_D) neL[tid] = ne[i * NB_D + tid];
    __syncthreads();
    const float stay = (float)(*stay_cost);

    // pass 1: logits (relu dot, diag override) + row max
    float lmax = -1e30f;
    for (int j = tid; j < NB_N; j += 256) {
        float v = 0.f;
#pragma unroll
        for (int d = 0; d < NB_D; ++d) v += neL[d] * ne[j * NB_D + d];
        v = fmaxf(v, 0.f);
        if (j == i) v = stay;
        S[(long)i * NB_N + j] = v;
        lmax = fmaxf(lmax, v);
    }
    red[tid] = lmax; __syncthreads();
    for (int s = 128; s > 0; s >>= 1) {
        if (tid < s) red[tid] = fmaxf(red[tid], red[tid + s]);
        __syncthreads();
    }
    const float rmax = red[0];
    __syncthreads();

    // pass 2: exp + row sum
    float lsum = 0.f;
    for (int j = tid; j < NB_N; j += 256) {
        const float e = __expf(S[(long)i * NB_N + j] - rmax);
        S[(long)i * NB_N + j] = e;
        lsum += e;
    }
    red[tid] = lsum; __syncthreads();
    for (int s = 128; s > 0; s >>= 1) {
        if (tid < s) red[tid] += red[tid + s];
        __syncthreads();
    }
    const float inv = 1.f / red[0];

    // pass 3: normalize
    for (int j = tid; j < NB_N; j += 256)
        S[(long)i * NB_N + j] *= inv;
}

// Pack x (B,N,C) -> Xp (N, B*C) so diffusion is one GEMM per hop.
__global__ void k_pack_x(const float* __restrict__ x, float* __restrict__ Xp)
{
    const long idx = (long)blockIdx.x * 256 + threadIdx.x;
    if (idx >= (long)NB_N * NB_BC) return;
    const int n = (int)(idx / NB_BC);
    const int col = (int)(idx % NB_BC);
    const int b = col / NB_C, c = col % NB_C;
    Xp[idx] = x[((long)b * NB_N + n) * NB_C + c];
}

// ---------------------------------------------------------------------------
// Diffusion-conv branch: out[:, :, 0:64).  Per-node workgroup; the adaptive
// weights (51x64, rank-16 in ne) are materialized in LDS and reused across
// all 16 batches.
// ---------------------------------------------------------------------------
__global__ __launch_bounds__(256) void k_sconv(
    const float* __restrict__ x, const float* __restrict__ Y1,
    const float* __restrict__ Y2, const float* __restrict__ ne,
    const float* __restrict__ wp, float* __restrict__ out)
{
    __shared__ float neL[NB_D];
    __shared__ float wL[NB_KC * NB_O4];   // 3264 floats
    __shared__ float xsL[NB_B * NB_KC];   // 816 floats
    const int n = blockIdx.x, tid = threadIdx.x;
    if (tid < NB_D) neL[tid] = ne[n * NB_D + tid];
    __syncthreads();

    for (int idx = tid; idx < NB_KC * NB_O4; idx += 256) {
        const int o = idx & 63;
        const int r = idx >> 6;           // r = k*17 + i
        float w = 0.f;
#pragma unroll
        for (int d = 0; d < NB_D; ++d)
            w += neL[d] * wp[((long)d * NB_KC + r) * NB_O4 + o];
        wL[idx] = w;
    }
    for (int idx = tid; idx < NB_B * NB_KC; idx += 256) {
        const int b = idx / NB_KC, r = idx % NB_KC;
        const int k = r / NB_C, i = r % NB_C;
        float v;
        if (k == 0)      v = x [((long)b * NB_N + n) * NB_C + i];
        else if (k == 1) v = Y1[(long)n * NB_BC + b * NB_C + i];
        else             v = Y2[(long)n * NB_BC + b * NB_C + i];
        xsL[idx] = v;
    }
    __syncthreads();

    for (int idx = tid; idx < NB_B * NB_O4; idx += 256) {
        const int b = idx >> 6, o = idx & 63;
        float acc = 0.f;
        for (int r = 0; r < NB_KC; ++r)
            acc += xsL[b * NB_KC + r] * wL[r * NB_O4 + o];
        out[((long)b * NB_N + n) * NB_O + o] = acc;
    }
}

// Supra branch step 1: bootstrap-gather + Linear(1,1); rows packed (q*16+b).
__global__ void k_xe_pack(const float* __restrict__ xew, const int* __restrict__ bidx,
                          const float* __restrict__ lw, const float* __restrict__ lb,
                          float* __restrict__ XeP)
{
    const long idx = (long)blockIdx.x * 256 + threadIdx.x;
    if (idx >= 48L * NB_E) return;
    const int row = (int)(idx / NB_E), i = (int)(idx % NB_E);
    const int q = row >> 4, b = row & 15;
    const int t = bidx[q];
    XeP[idx] = xew[((long)(b * NB_W + t)) * NB_E + i] * lw[0] + lb[0];
}

// Supra step 2 epilogue: add strict-upper-block jump_cost contributions.
__global__ void k_xes_combine(const float* __restrict__ Hout,
                              const float* __restrict__ XeP,
                              const int* __restrict__ jump,
                              float* __restrict__ Xes)
{
    const long idx = (long)blockIdx.x * 256 + threadIdx.x;
    if (idx >= 48L * NB_E) return;
    const int row = (int)(idx / NB_E), j = (int)(idx % NB_E);
    const int q = row >> 4, b = row & 15;
    const float jc = (float)(*jump);
    float acc = Hout[idx];
    for (int p = 0; p < q; ++p)
        acc += jc * XeP[((long)(p * 16 + b)) * NB_E + j];
    Xes[idx] = acc;
}

// Supra branch output: out[:, :, 224:256) = mean_f(xen) * (ne @ wws_p)
__global__ __launch_bounds__(256) void k_supra_out(
    const float* __restrict__ Xen, const float* __restrict__ ne,
    const float* __restrict__ wws_p, float* __restrict__ out)
{
    __shared__ float neL[NB_D];
    __shared__ float wwsL[32];
    const int n = blockIdx.x, tid = threadIdx.x;
    if (tid < NB_D) neL[tid] = ne[n * NB_D + tid];
    __syncthreads();
    if (tid < 32) {
        float w = 0.f;
#pragma unroll
        for (int d = 0; d < NB_D; ++d) w += neL[d] * wws_p[d * 32 + tid];
        wwsL[tid] = w;
    }
    __syncthreads();
    for (int idx = tid; idx < NB_B * 32; idx += 256) {
        const int b = idx >> 5, o = idx & 31;
        const float s = (Xen[(long)b * NB_N + n] +
                         Xen[(long)(16 + b) * NB_N + n] +
                         Xen[(long)(32 + b) * NB_N + n]) * (1.f / 3.f);
        out[((long)b * NB_N + n) * NB_O + 224 + o] = s * wwsL[o];
    }
}

// Temporal branch: out[:, :, 96:224) = (sum_t xw*T) * (ne @ wwt_p)
__global__ __launch_bounds__(256) void k_temporal(
    const float* __restrict__ xw, const float* __restrict__ ne,
    const float* __restrict__ wwt_p, const float* __restrict__ Tp,
    float* __restrict__ out)
{
    __shared__ float neL[NB_D];
    __shared__ float wwtL[128];
    __shared__ float sL[NB_B];
    const int n = blockIdx.x, tid = threadIdx.x;
    if (tid < NB_D) neL[tid] = ne[n * NB_D + tid];
    __syncthreads();
    if (tid < 128) {
        float w = 0.f;
#pragma unroll
        for (int d = 0; d < NB_D; ++d) w += neL[d] * wwt_p[d * 128 + tid];
        wwtL[tid] = w;
    } else if (tid < 144) {
        const int b = tid - 128;
        float s = 0.f;
#pragma unroll
        for (int t = 0; t < NB_W; ++t)
            s += xw[((long)(b * NB_W + t)) * NB_N + n] * Tp[t];
        sL[b] = s;
    }
    __syncthreads();
    for (int idx = tid; idx < NB_B * 128; idx += 256) {
        const int b = idx >> 7, o = idx & 127;
        out[((long)b * NB_N + n) * NB_O + 96 + o] = sL[b] * wwtL[o];
    }
}

// ZFC branch: gnn_w row sums (ZFC is constant along the contracted axis).
__global__ __launch_bounds__(256) void k_gsum(const float* __restrict__ gw,
                                              float* __restrict__ gsum)
{
    __shared__ float red[256];
    const int n = blockIdx.x, tid = threadIdx.x;
    float s = 0.f;
    for (int m = tid; m < NB_N; m += 256) s += gw[(long)n * NB_N + m];
    red[tid] = s; __syncthreads();
    for (int k = 128; k > 0; k >>= 1) {
        if (tid < k) red[tid] += red[tid + k];
        __syncthreads();
    }
    if (tid == 0) gsum[n] = red[0];
}

// ZFC branch: relu + the (B,32,N)->(B,N,32) *reshape* (index math, no transpose).
__global__ void k_zfc(const float* __restrict__ zin, const float* __restrict__ gsum,
                      const float* __restrict__ gb, float* __restrict__ out)
{
    const long idx = (long)blockIdx.x * 256 + threadIdx.x;
    if (idx >= (long)NB_B * NB_N * 32) return;
    const int b = (int)(idx / (NB_N * 32));
    const int r = (int)(idx % (NB_N * 32));     // flat index = n*32 + c
    const int n = r >> 5, c = r & 31;
    const int o = r / NB_N, m = r % NB_N;       // source position in (32, N)
    const float z = zin[b * 32 + o] * gsum[m] + gb[m];
    out[((long)b * NB_N + n) * NB_O + 64 + c] = fmaxf(z, 0.f);
}

// Final: out += ne @ bias_pool (runs last, accumulates over all 256 columns).
__global__ __launch_bounds__(256) void k_bias(
    const float* __restrict__ ne, const float* __restrict__ bp,
    float* __restrict__ out)
{
    __shared__ float neL[NB_D];
    const int n = blockIdx.x, o = threadIdx.x;
    if (o < NB_D) neL[o] = ne[n * NB_D + o];
    __syncthreads();
    float bsum = 0.f;
#pragma unroll
    for (int d = 0; d < NB_D; ++d) bsum += neL[d] * bp[d * NB_O + o];
    for (int b = 0; b < NB_B; ++b)
        out[((long)b * NB_N + n) * NB_O + o] += bsum;
}

// ---------------------------------------------------------------------------
extern "C" void kernel_launch(void* const* d_in, const int* in_sizes, int n_in,
                              void* d_out, int out_size, void* d_ws, size_t ws_size,
                              hipStream_t stream) {
    const float* x     = (const float*)d_in[0];
    const float* xw    = (const float*)d_in[1];
    const float* ne    = (const float*)d_in[2];
    // d_in[3] fixed_adj == 0 -> softmax path (static in reference)
    // d_in[4] adj unused on this path
    const int*   stay  = (const int*)d_in[5];
    const int*   jump  = (const int*)d_in[6];
    const float* zin   = (const float*)d_in[7];
    const float* hodge = (const float*)d_in[8];
    const float* xew   = (const float*)d_in[9];
    const float* inc   = (const float*)d_in[10];
    const float* wp    = (const float*)d_in[11];
    const float* wwsp  = (const float*)d_in[12];
    const float* wwtp  = (const float*)d_in[13];
    const float* bp    = (const float*)d_in[14];
    const float* Tp    = (const float*)d_in[15];
    const float* lw    = (const float*)d_in[16];
    const float* lb    = (const float*)d_in[17];
    const float* gw    = (const float*)d_in[18];
    const float* gb    = (const float*)d_in[19];
    const int*   bidx  = (const int*)d_in[20];
    float* out = (float*)d_out;

    // Workspace carve-up (floats). Total ~6.17M floats ~ 24.7 MB.
    float* ws   = (float*)d_ws;
    float* S    = ws;                       // N*N      = 4,000,000
    float* Xp   = S    + (long)NB_N * NB_N; // N*BC     =   544,000
    float* Y1   = Xp   + (long)NB_N * NB_BC;
    float* Y2   = Y1   + (long)NB_N * NB_BC;
    float* XeP  = Y2   + (long)NB_N * NB_BC; // 48*E    =   144,000
    float* Hout = XeP  + 48L * NB_E;
    float* Xes  = Hout + 48L * NB_E;
    float* Xen  = Xes  + 48L * NB_E;         // 48*N    =    96,000
    float* gsum = Xen  + 48L * NB_N;         // N

    // 1) adaptive transition matrix S
    k_compute_S<<<NB_N, 256, 0, stream>>>(ne, stay, S);

    // 2) pack x and run the two diffusion hops as WMMA GEMMs:
    //    Y1 = S @ Xp   (2000 x 272 x 2000),  Y2 = S @ Y1  (== S^2 @ Xp)
    {
        const long tot = (long)NB_N * NB_BC;
        k_pack_x<<<(unsigned)((tot + 255) / 256), 256, 0, stream>>>(x, Xp);
    }
    k_wmma_gemm_f32<<<dim3(NB_N / 16, NB_BC / 16), 32, 0, stream>>>(
        S, NB_N, Xp, NB_BC, 1, Y1, NB_BC, NB_N, NB_BC, NB_N);
    k_wmma_gemm_f32<<<dim3(NB_N / 16, NB_BC / 16), 32, 0, stream>>>(
        S, NB_N, Y1, NB_BC, 1, Y2, NB_BC, NB_N, NB_BC, NB_N);

    // 3) diffusion-conv branch -> out[:, :, 0:64)
    k_sconv<<<NB_N, 256, 0, stream>>>(x, Y1, Y2, ne, wp, out);

    // 4) supra branch (block-structured supra-Laplacian):
    //    Hout = XeP @ hodge (48 x 3000 x 3000), epilogue adds jump terms,
    //    Xen  = Xes @ inc^T (48 x 2000 x 3000)
    {
        const long tot = 48L * NB_E;
        k_xe_pack<<<(unsigned)((tot + 255) / 256), 256, 0, stream>>>(xew, bidx, lw, lb, XeP);
        k_wmma_gemm_f32<<<dim3(48 / 16, (NB_E + 15) / 16), 32, 0, stream>>>(
            XeP, NB_E, hodge, NB_E, 1, Hout, NB_E, 48, NB_E, NB_E);
        k_xes_combine<<<(unsigned)((tot + 255) / 256), 256, 0, stream>>>(Hout, XeP, jump, Xes);
        k_wmma_gemm_f32<<<dim3(48 / 16, NB_N / 16), 32, 0, stream>>>(
            Xes, NB_E, inc, 1, NB_E, Xen, NB_N, 48, NB_N, NB_E);
        k_supra_out<<<NB_N, 256, 0, stream>>>(Xen, ne, wwsp, out);
    }

    // 5) temporal branch -> out[:, :, 96:224)
    k_temporal<<<NB_N, 256, 0, stream>>>(xw, ne, wwtp, Tp, out);

    // 6) ZFC branch -> out[:, :, 64:96)
    k_gsum<<<NB_N, 256, 0, stream>>>(gw, gsum);
    {
        const long tot = (long)NB_B * NB_N * 32;
        k_zfc<<<(unsigned)((tot + 255) / 256), 256, 0, stream>>>(zin, gsum, gb, out);
    }

    // 7) bias add (last; every output column already written exactly once)
    k_bias<<<NB_N, 256, 0, stream>>>(ne, bp, out);
}